// StructuralFingerprintLayer_74122545594651
// MI455X (gfx1250) — compile-verified
//
#include <hip/hip_runtime.h>
#include <hip/hip_bf16.h>
#include <math.h>

// ---------------------------------------------------------------------------
// GAT layer on MI455X (gfx1250), single-pass online softmax + fp32 WMMA.
//   h = input @ W                      (8192x512)@(512x64)  -> WMMA f32 16x16x4
//   e = |W_ei|*leaky(s1[i]+s2[j]) + |W_si|*adj_ad ; mask by adj>0
//   out = elu(softmax(e) @ h)          flash-style single pass over 512 MB
// HBM floor: 512 MB @ 23.3 TB/s ~ 22us ; attention GEMM needs 16 FLOP/byte,
// far below the f32 matrix-core ceiling, so fp32 WMMA is the right precision.
// ---------------------------------------------------------------------------

typedef float v2f __attribute__((ext_vector_type(2)));
typedef float v8f __attribute__((ext_vector_type(8)));

#define NN      8192
#define IN_F    512
#define OUT_F   64
#define LALPHA  0.2f
#define NEG_BIG -9000000000000000.0f

// ---------------------------------------------------------------------------
// Kernel 1: one wave per 16-row tile of h.
//   h(16x64) = input(16x512) @ W(512x64) via 16x16x4 f32 WMMA (K-steps of 4).
//   Fused: s1 = h@a1, s2 = h@a2 reduced from the register-resident tile.
// A layout (16x4 f32): lane L<16 -> M=L, K={k,k+1}; lane 16+L -> M=L, K={k+2,k+3}
// B layout (4x16 f32): mirrored; C/D: VGPR r -> rows r (lanes 0-15) / r+8 (16-31)
// ---------------------------------------------------------------------------
__global__ __launch_bounds__(32) void gat_hproj_kernel(
    const float* __restrict__ input, const float* __restrict__ W,
    const float* __restrict__ a,
    float* __restrict__ h, float* __restrict__ s1, float* __restrict__ s2)
{
    const int lane = threadIdx.x & 31;
    const int half = lane >> 4;     // 0: K pair {0,1}; 1: K pair {2,3}
    const int l16  = lane & 15;
    const int i0   = blockIdx.x * 16;

    v8f acc[4] = {};                // 16x64 output tile, 4 column blocks

    for (int k = 0; k < IN_F; k += 4) {
        // A: input rows i0+l16, this lane's K pair
        const float* ap = input + (size_t)(i0 + l16) * IN_F + k + 2 * half;
        v2f A; A.x = ap[0]; A.y = ap[1];
#pragma unroll
        for (int c = 0; c < 4; ++c) {
            const float* bp = W + (size_t)(k + 2 * half) * OUT_F + c * 16 + l16;
            v2f B; B.x = bp[0]; B.y = bp[OUT_F];
            acc[c] = __builtin_amdgcn_wmma_f32_16x16x4_f32(
                false, A, false, B, (short)0, acc[c], false, false);
        }
    }

    // Store h tile: element (c,r) is row i0+r+8*half, col 16c+l16
#pragma unroll
    for (int c = 0; c < 4; ++c)
#pragma unroll
        for (int r = 0; r < 8; ++r)
            h[(size_t)(i0 + r + 8 * half) * OUT_F + c * 16 + l16] = acc[c][r];

    // Fused s1/s2: per row dot with a1/a2, reduce across the 16-lane half
    const float* a1 = a;
    const float* a2 = a + OUT_F;
#pragma unroll
    for (int r = 0; r < 8; ++r) {
        float d1 = 0.f, d2 = 0.f;
#pragma unroll
        for (int c = 0; c < 4; ++c) {
            float hv = acc[c][r];
            d1 += hv * a1[c * 16 + l16];
            d2 += hv * a2[c * 16 + l16];
        }
#pragma unroll
        for (int off = 8; off >= 1; off >>= 1) {
            d1 += __shfl_xor(d1, off, 32);
            d2 += __shfl_xor(d2, off, 32);
        }
        if (l16 == 0) {
            const int row = i0 + r + 8 * half;
            s1[row] = d1;
            s2[row] = d2;
        }
    }
}

// ---------------------------------------------------------------------------
// Kernel 2: one wave per 16-row block; online softmax over 512 column tiles.
// Attention logits are produced directly in WMMA A-layout:
//   lane L<16  -> row L,  cols j0+{4q, 4q+1}
//   lane 16+L  -> row L,  cols j0+{4q+2, 4q+3}
// Row state (m, l) replicated in lanes L and L+16 (synced via shfl_xor 16).
// acc (16x64, C-layout) rescaled per tile via ds_bpermute row broadcasts.
// ---------------------------------------------------------------------------
__global__ __launch_bounds__(32) void gat_attn_kernel(
    const float* __restrict__ adj_ad, const int* __restrict__ adj,
    const float* __restrict__ h,
    const float* __restrict__ s1v, const float* __restrict__ s2v,
    const float* __restrict__ W_si, const float* __restrict__ W_ei,
    float* __restrict__ out)
{
    const int lane = threadIdx.x & 31;
    const int half = lane >> 4;
    const int l16  = lane & 15;
    const int i0   = blockIdx.x * 16;
    const int row  = i0 + l16;

    const float aWei = fabsf(W_ei[0]);
    const float aWsi = fabsf(W_si[0]);
    const float s1   = s1v[row];

    const float* adrow = adj_ad + (size_t)row * NN;
    const int*   ajrow = adj    + (size_t)row * NN;

    float m = -INFINITY;    // running row max
    float l = 0.f;          // running row sum
    v8f acc[4] = {};        // running weighted sum (16x64, C-layout)

    for (int j0 = 0; j0 < NN; j0 += 16) {
        // speculative prefetch a few tiles ahead (adjacency streams from HBM)
        if (j0 + 64 < NN) {
            __builtin_prefetch(adrow + j0 + 64, 0, 0);
            __builtin_prefetch(ajrow + j0 + 64, 0, 0);
        }

        // ---- logits, directly in A-layout (8 per lane as 4 pairs) ----
        float p[4][2];
        float tmax = -INFINITY;
#pragma unroll
        for (int q = 0; q < 4; ++q) {
            const int col = j0 + 4 * q + 2 * half;
            const float2 s2p = *(const float2*)(s2v   + col);
            const float2 adp = *(const float2*)(adrow + col);
            const int2   ajp = *(const int2*)  (ajrow + col);
#pragma unroll
            for (int t = 0; t < 2; ++t) {
                float x = s1 + (t ? s2p.y : s2p.x);
                x = (x > 0.f) ? x : LALPHA * x;                    // leaky_relu
                float e = aWei * x + aWsi * (t ? adp.y : adp.x);
                e = ((t ? ajp.y : ajp.x) > 0) ? e : NEG_BIG;       // mask
                p[q][t] = e;
                tmax = fmaxf(tmax, e);
            }
        }
        tmax = fmaxf(tmax, __shfl_xor(tmax, 16, 32));   // full row max

        // ---- online softmax update ----
        const float newm  = fmaxf(m, tmax);
        const float scale = __expf(m - newm);
        float rsum = 0.f;
#pragma unroll
        for (int q = 0; q < 4; ++q)
#pragma unroll
            for (int t = 0; t < 2; ++t) {
                const float pe = __expf(p[q][t] - newm);
                p[q][t] = pe;
                rsum += pe;
            }
        rsum += __shfl_xor(rsum, 16, 32);
        l = l * scale + rsum;
        m = newm;

        // rescale accumulators: C-layout VGPR r holds row r (lanes<16) / r+8
#pragma unroll
        for (int r = 0; r < 8; ++r) {
            const float sc = __shfl(scale, half ? (r + 8) : r, 32);
#pragma unroll
            for (int c = 0; c < 4; ++c) acc[c][r] *= sc;
        }

        // ---- acc += P(16x16) @ H(16x64) as 16 x (16x16x4 f32 WMMA) ----
#pragma unroll
        for (int q = 0; q < 4; ++q) {
            v2f A; A.x = p[q][0]; A.y = p[q][1];
            const float* hb = h + (size_t)(j0 + 4 * q + 2 * half) * OUT_F;
#pragma unroll
            for (int c = 0; c < 4; ++c) {
                v2f B; B.x = hb[c * 16 + l16]; B.y = hb[OUT_F + c * 16 + l16];
                acc[c] = __builtin_amdgcn_wmma_f32_16x16x4_f32(
                    false, A, false, B, (short)0, acc[c], false, false);
            }
        }
    }

    // ---- finalize: normalize, elu, store ----
#pragma unroll
    for (int r = 0; r < 8; ++r) {
        const float li  = __shfl(l, half ? (r + 8) : r, 32);
        const float inv = 1.0f / li;
#pragma unroll
        for (int c = 0; c < 4; ++c) {
            float v = acc[c][r] * inv;
            v = (v > 0.f) ? v : (__expf(v) - 1.0f);                // elu
            out[(size_t)(i0 + r + 8 * half) * OUT_F + c * 16 + l16] = v;
        }
    }
}

// ---------------------------------------------------------------------------
extern "C" void kernel_launch(void* const* d_in, const int* in_sizes, int n_in,
                              void* d_out, int out_size, void* d_ws, size_t ws_size,
                              hipStream_t stream) {
    (void)in_sizes; (void)n_in; (void)out_size; (void)ws_size;

    const float* input  = (const float*)d_in[0];   // (8192, 512)
    const float* W      = (const float*)d_in[1];   // (512, 64)
    const float* a      = (const float*)d_in[2];   // (128, 1)
    const float* W_si   = (const float*)d_in[3];   // (1, 1)
    const float* W_ei   = (const float*)d_in[4];   // (1, 1)
    const float* adj_ad = (const float*)d_in[5];   // (8192, 8192) f32
    const int*   adj    = (const int*)  d_in[6];   // (8192, 8192) i32
    float* out = (float*)d_out;                    // (8192, 64) f32

    // workspace: h (8192*64 f32 = 2 MB) + s1/s2 (32 KB each)
    float* h  = (float*)d_ws;
    float* s1 = h + (size_t)NN * OUT_F;
    float* s2 = s1 + NN;

    gat_hproj_kernel<<<NN / 16, 32, 0, stream>>>(input, W, a, h, s1, s2);
    gat_attn_kernel<<<NN / 16, 32, 0, stream>>>(adj_ad, adj, h, s1, s2,
                                                W_si, W_ei, out);
}